// renderModule_31404800869036
// MI455X (gfx1250) — compile-verified
//
#include <hip/hip_runtime.h>
#include <math.h>

// ---------------------------------------------------------------------------
// Point-splat renderer for MI455X (gfx1250, wave32).
//   pass 0: zero accumulators (d_out = rgb acc, d_ws = weight acc)
//   pass 1: WMMA-based batched affine transform + perspective + float-atomic
//           scatter (global_atomic_add_f32; accumulator is L2-resident)
//   pass 2: fused normalize + background blend, in place on d_out
// ---------------------------------------------------------------------------

typedef float v2f __attribute__((ext_vector_type(2)));
typedef float v8f __attribute__((ext_vector_type(8)));

#define IMG_W 1024
#define IMG_H 1024
#define BG_WF 0.001f
#define EPSF  1e-6f

__global__ void zero_f32_kernel(float* __restrict__ a, int n) {
  for (int i = blockIdx.x * blockDim.x + threadIdx.x; i < n;
       i += gridDim.x * blockDim.x)
    a[i] = 0.0f;
}

// One wave handles 32 points per iteration: two V_WMMA_F32_16X16X4_F32 ops
// compute pc = [x y z 1] * [R^T ; t] for 16 points each; the 16x16 D tile's
// first 3 columns are bounced through LDS to get one point per lane.
__global__ __launch_bounds__(256)
void splat_kernel(const float* __restrict__ pts,   // N x 6 (xyz rgb)
                  const float* __restrict__ cam,   // 16 floats
                  float* __restrict__ acc,         // H*W*3 rgb accumulator
                  float* __restrict__ wsum,        // H*W weight accumulator
                  int npts)
{
  __shared__ float lds[8 * 96];            // 96 floats per wave (2 tiles x 3x16)
  const int lane = threadIdx.x & 31;
  const int wave = threadIdx.x >> 5;
  float* wl = &lds[wave * 96];

  const float fx = cam[12], fy = cam[13], cx = cam[14], cy = cam[15];

  // B operand (4x16, K x N): column n<3 is [R[n][0], R[n][1], R[n][2], t[n]].
  // Row-striped layout: VGPR0 = rows K0 (lanes 0-15) / K2 (lanes 16-31),
  //                     VGPR1 = rows K1 (lanes 0-15) / K3 (lanes 16-31).
  const int  n  = lane & 15;
  const bool hi = lane >= 16;
  v2f B = {0.0f, 0.0f};
  if (n < 3) {
    if (!hi) { B.x = cam[n * 3 + 0]; B.y = cam[n * 3 + 1]; }  // R[n][0], R[n][1]
    else     { B.x = cam[n * 3 + 2]; B.y = cam[9 + n];      } // R[n][2], t[n]
  }

  // Per-lane fixed part of the A-operand address: lanes 0-15 read (x,y),
  // lanes 16-31 read (z, <ignored>) -- one aligned b64 load per lane.
  const int laneOff = n * 6 + (hi ? 2 : 0);

  const int tiles  = (npts + 31) >> 5;     // 32-point tiles
  const int stride = gridDim.x * 8;        // 8 waves per 256-thread block

  for (int tile = blockIdx.x * 8 + wave; tile < tiles; tile += stride) {
    const int base = tile << 5;

    // Prefetch next tile's point data (global_prefetch_b8).
    {
      const int nb = base + stride * 32;
      if (nb < npts)
        __builtin_prefetch(pts + (size_t)(nb + lane) * 6, 0, 3);
    }

    #pragma unroll
    for (int half = 0; half < 2; ++half) {
      // A operand (16x4): row m = [x y z 1] of point base+half*16+m.
      // Lane m holds K0,K1 = (x,y); lane m+16 holds K2,K3 = (z,1).
      const int p = base + half * 16 + n;
      v2f A = {0.0f, 0.0f};
      if (p < npts) {
        const v2f v = *(const v2f*)(pts + (size_t)(base + half * 16) * 6 + laneOff);
        A.x = v.x;
        A.y = hi ? 1.0f : v.y;             // v_cndmask, no second load
      }
      v8f C = {};
      v8f D = __builtin_amdgcn_wmma_f32_16x16x4_f32(
          /*neg_a=*/false, A, /*neg_b=*/false, B,
          /*c_mod=*/(short)0, C, /*reuse_a=*/false, /*reuse_b=*/false);

      // D layout: VGPR r = row r (lanes 0-15) / row r+8 (lanes 16-31),
      // column = lane&15. Keep columns 0..2 (= pc.x/y/z), store column-major.
      if (n < 3) {
        float* dst = wl + half * 48 + n * 16 + (hi ? 8 : 0);
        #pragma unroll
        for (int r = 0; r < 8; ++r) dst[r] = D[r];
      }
    }
    __builtin_amdgcn_wave_barrier();       // order same-wave LDS store -> load

    const int t2 = lane >> 4, m = lane & 15;
    const float pcx = wl[t2 * 48 +  0 + m];
    const float pcy = wl[t2 * 48 + 16 + m];
    const float pcz = wl[t2 * 48 + 32 + m];
    __builtin_amdgcn_wave_barrier();

    const int p = base + lane;
    if (p < npts) {
      const float zs  = fmaxf(pcz, EPSF);
      const float inv = 1.0f / zs;
      const float u = fx * pcx * inv + cx;
      const float v = fy * pcy * inv + cy;
      const int ui = (int)floorf(u);
      const int vi = (int)floorf(v);
      if (pcz > EPSF && ui >= 0 && ui < IMG_W && vi >= 0 && vi < IMG_H) {
        const size_t p6 = (size_t)p * 6;
        const float cr = pts[p6 + 3], cg = pts[p6 + 4], cb = pts[p6 + 5];
        const int idx = vi * IMG_W + ui;
        unsafeAtomicAdd(&acc[(size_t)idx * 3 + 0], inv * cr);
        unsafeAtomicAdd(&acc[(size_t)idx * 3 + 1], inv * cg);
        unsafeAtomicAdd(&acc[(size_t)idx * 3 + 2], inv * cb);
        unsafeAtomicAdd(&wsum[idx], inv);
      }
    }
  }
}

__global__ void finalize_kernel(float* __restrict__ out,
                                const float* __restrict__ wsum,
                                const float* __restrict__ env, int hw) {
  for (int i = blockIdx.x * blockDim.x + threadIdx.x; i < hw;
       i += gridDim.x * blockDim.x) {
    const float inv = 1.0f / (wsum[i] + BG_WF);
    const size_t b = (size_t)i * 3;
    out[b + 0] = (out[b + 0] + BG_WF * env[b + 0]) * inv;
    out[b + 1] = (out[b + 1] + BG_WF * env[b + 1]) * inv;
    out[b + 2] = (out[b + 2] + BG_WF * env[b + 2]) * inv;
  }
}

extern "C" void kernel_launch(void* const* d_in, const int* in_sizes, int n_in,
                              void* d_out, int out_size, void* d_ws, size_t ws_size,
                              hipStream_t stream) {
  // inputs: [0] cam_type(int,1)  [1] camera(16)  [2] points(N*6)  [3] env(H*W*3)
  const float* cam = (const float*)d_in[1];
  const float* pts = (const float*)d_in[2];
  const float* env = (const float*)d_in[3];
  float* out  = (float*)d_out;             // rgb accumulator, then final image
  float* wsum = (float*)d_ws;              // H*W floats of scratch (4 MB)

  const int npts = in_sizes[2] / 6;
  const int hw   = in_sizes[3] / 3;        // 1024*1024

  // pass 0: zero accumulators (harness does not re-poison between replays)
  zero_f32_kernel<<<2048, 256, 0, stream>>>(out, hw * 3);
  zero_f32_kernel<<<1024, 256, 0, stream>>>(wsum, hw);

  // pass 1: transform + scatter
  const int tiles  = (npts + 31) / 32;
  int blocks = (tiles + 7) / 8;
  if (blocks > 8192) blocks = 8192;
  if (blocks < 1)    blocks = 1;
  splat_kernel<<<blocks, 256, 0, stream>>>(pts, cam, out, wsum, npts);

  // pass 2: normalize + background blend
  finalize_kernel<<<2048, 256, 0, stream>>>(out, wsum, env, hw);
}